// CDFLoss_60155311948183
// MI455X (gfx1250) — compile-verified
//
#include <hip/hip_runtime.h>
#include <stdint.h>

// Problem constants (match reference)
#define B_DIM 32768
#define T_DIM 1024
#define ROWS_PER_BLOCK 8
#define THREADS 256
#define EPSF 1e-7f
#define MARGINF 0.1f

// ---------------------------------------------------------------------------
// Zero the two f64 accumulators (bce_sum, mono_sum) in workspace.
// ---------------------------------------------------------------------------
__global__ __launch_bounds__(64) void cdf_init(double* __restrict__ acc) {
    if (threadIdx.x < 2) acc[threadIdx.x] = 0.0;
}

// ---------------------------------------------------------------------------
// Main streaming kernel: each block handles 8 contiguous rows of F_pred.
// Rows are staged in LDS via the CDNA5 async copy engine
// (global_load_async_to_lds_b128, ASYNCcnt), double-buffered so the next
// row's HBM fetch overlaps this row's log/relu math. Neighbor access for the
// monotonic term comes from LDS, so each F_pred byte is read from HBM once.
// ---------------------------------------------------------------------------
__global__ __launch_bounds__(THREADS) void cdf_main(const float* __restrict__ F,
                                                    const int*   __restrict__ dur,
                                                    const int*   __restrict__ ev,
                                                    double*      __restrict__ acc) {
    __shared__ __align__(16) float buf[2][T_DIM];

    const int      r0   = blockIdx.x * ROWS_PER_BLOCK;
    const int      tid  = threadIdx.x;
    const uint32_t voff = (uint32_t)(tid * 16);   // 16 B (float4) per lane

    // Preload row 0 into buffer 0 (one b128 async load per lane covers 4 KB/row)
    {
        uint32_t dst = (uint32_t)(uintptr_t)(&buf[0][tid * 4]);
        unsigned long long src =
            (unsigned long long)(uintptr_t)(F + (size_t)r0 * T_DIM);
        asm volatile("global_load_async_to_lds_b128 %0, %1, %2"
                     :: "v"(dst), "v"(voff), "s"(src)
                     : "memory");
    }

    float bce_acc  = 0.0f;
    float mono_acc = 0.0f;

    for (int i = 0; i < ROWS_PER_BLOCK; ++i) {
        const int row = r0 + i;

        if (i + 1 < ROWS_PER_BLOCK) {
            // Kick off next row into the other buffer, then wait for current.
            uint32_t dst = (uint32_t)(uintptr_t)(&buf[(i + 1) & 1][tid * 4]);
            unsigned long long src =
                (unsigned long long)(uintptr_t)(F + (size_t)(row + 1) * T_DIM);
            asm volatile("global_load_async_to_lds_b128 %0, %1, %2"
                         :: "v"(dst), "v"(voff), "s"(src)
                         : "memory");
            asm volatile("s_wait_asynccnt 0x1" ::: "memory");
        } else {
            asm volatile("s_wait_asynccnt 0x0" ::: "memory");
        }
        __syncthreads();   // entire row now visible in LDS to all 8 waves

        const float* b  = buf[i & 1];
        const int    t0 = tid * 4;
        const float4 f  = *reinterpret_cast<const float4*>(b + t0);

        const int d = dur[row];   // uniform per row -> scalar loads
        const int e = ev[row];

        // --- monotonic term: relu(F[t] - F[t+1] + margin), t in [0, T-2] ---
        mono_acc += fmaxf(f.x - f.y + MARGINF, 0.0f);
        mono_acc += fmaxf(f.y - f.z + MARGINF, 0.0f);
        mono_acc += fmaxf(f.z - f.w + MARGINF, 0.0f);
        if (tid < THREADS - 1) {
            const float fn = b[t0 + 4];     // neighbor from LDS, not HBM
            mono_acc += fmaxf(f.w - fn + MARGINF, 0.0f);
        }

        // --- masked BCE: exactly one v_log_f32 per masked element ---
        const float v[4] = {f.x, f.y, f.z, f.w};
#pragma unroll
        for (int k = 0; k < 4; ++k) {
            const int  t   = t0 + k;
            const bool msk = e ? true : (t <= d);
            const bool tgt = e ? (t >= d) : false;
            const float p  = fminf(fmaxf(v[k], EPSF), 1.0f - EPSF);
            const float q  = tgt ? p : (1.0f - p);   // log(p) or log(1-p)
            const float l  = __logf(q);              // v_log_f32 path
            bce_acc += msk ? -l : 0.0f;
        }
        __syncthreads();   // buffer (i&1) may be overwritten next iteration
    }

    // --- block reduction: wave32 shuffle tree, then cross-wave via LDS ---
#pragma unroll
    for (int off = 16; off > 0; off >>= 1) {
        bce_acc  += __shfl_down(bce_acc,  off, 32);
        mono_acc += __shfl_down(mono_acc, off, 32);
    }
    __shared__ float rb[THREADS / 32];
    __shared__ float rm[THREADS / 32];
    const int wid  = tid >> 5;
    const int lane = tid & 31;
    if (lane == 0) { rb[wid] = bce_acc; rm[wid] = mono_acc; }
    __syncthreads();
    if (tid == 0) {
        float sb = 0.0f, sm = 0.0f;
#pragma unroll
        for (int w = 0; w < THREADS / 32; ++w) { sb += rb[w]; sm += rm[w]; }
        atomicAdd(&acc[0], (double)sb);   // f64 accumulators: order-robust
        atomicAdd(&acc[1], (double)sm);
    }
}

// ---------------------------------------------------------------------------
// Finalize: analytic mask count (event ? T : duration+1), bias regularizer,
// combine everything in f64, emit f32 loss.
// ---------------------------------------------------------------------------
__global__ __launch_bounds__(1024) void cdf_final(const float*  __restrict__ biases,
                                                  const int*    __restrict__ dur,
                                                  const int*    __restrict__ ev,
                                                  const double* __restrict__ acc,
                                                  float*        __restrict__ out) {
    __shared__ float sb[1024];
    __shared__ int   sc[1024];
    const int tid = threadIdx.x;

    const float bv = biases[tid];
    int cnt = 0;
    for (int r = tid; r < B_DIM; r += 1024)
        cnt += ev[r] ? T_DIM : (dur[r] + 1);

    sb[tid] = bv * bv;
    sc[tid] = cnt;
    __syncthreads();

    for (int s = 512; s > 0; s >>= 1) {
        if (tid < s) { sb[tid] += sb[tid + s]; sc[tid] += sc[tid + s]; }
        __syncthreads();
    }

    if (tid == 0) {
        const double bce_sum   = acc[0];
        const double mono_sum  = acc[1];
        const double mask_cnt  = (double)sc[0];
        const double bias_mean = (double)sb[0] / (double)T_DIM;
        const double mono_mean = mono_sum / ((double)B_DIM * (double)(T_DIM - 1));
        const double loss = bce_sum / mask_cnt + 0.1 * mono_mean + 0.1 * bias_mean;
        out[0] = (float)loss;
    }
}

// ---------------------------------------------------------------------------
extern "C" void kernel_launch(void* const* d_in, const int* in_sizes, int n_in,
                              void* d_out, int out_size, void* d_ws, size_t ws_size,
                              hipStream_t stream) {
    (void)in_sizes; (void)n_in; (void)out_size; (void)ws_size;

    const float* F      = (const float*)d_in[0];   // F_pred  [B*T] f32
    const float* biases = (const float*)d_in[1];   // biases  [T]   f32
    const int*   dur    = (const int*)d_in[2];     // duration[B]   int
    const int*   ev     = (const int*)d_in[3];     // event   [B]   int
    float*       out    = (float*)d_out;           // scalar loss
    double*      acc    = (double*)d_ws;           // 2 x f64 accumulators

    cdf_init <<<1, 64, 0, stream>>>(acc);
    cdf_main <<<B_DIM / ROWS_PER_BLOCK, THREADS, 0, stream>>>(F, dur, ev, acc);
    cdf_final<<<1, 1024, 0, stream>>>(biases, dur, ev, acc, out);
}